// FixedFoveatedSensor_58566174049072
// MI455X (gfx1250) — compile-verified
//
#include <hip/hip_runtime.h>
#include <cstdint>

#define SENS 256
#define SPP_N 16
#define IMGD 1024
#define NBATCH 4
#define NCHAN 3
#define PIX_PER_BLK 64

// ---- gfx1250 hardware tanh (new CDNA5 transcendental) ----
__device__ __forceinline__ float gfx_tanh(float x) {
#if __has_builtin(__builtin_amdgcn_tanhf)
    return __builtin_amdgcn_tanhf(x);
#elif __has_builtin(__builtin_amdgcn_tanh_f32)
    return __builtin_amdgcn_tanh_f32(x);
#else
    float r;
    asm("v_tanh_f32 %0, %1" : "=v"(r) : "v"(x));
    return r;
#endif
}

// ---- gfx1250 async global->LDS DMA (no VGPR staging, tracked by ASYNCcnt) ----
// NT hint: jitter is single-use streaming data; keep the 192MB L2 for the image.
__device__ __forceinline__ void async_b128_to_lds_nt(uint32_t lds_byte_addr, uint64_t gaddr) {
    asm volatile("global_load_async_to_lds_b128 %0, %1, off th:TH_LOAD_NT"
                 :: "v"(lds_byte_addr), "v"(gaddr)
                 : "memory");
}
__device__ __forceinline__ void wait_asynccnt0() {
    asm volatile("s_wait_asynccnt 0" ::: "memory");
}

__global__ __launch_bounds__(256, 4)
void foveated_kernel(const float* __restrict__ img,
                     const float* __restrict__ t,
                     const float* __restrict__ jitter,
                     float* __restrict__ out) {
    // 8 KB: jitter[s][pix 0..63][xy] for this block's 64-pixel row segment
    __shared__ float ljit[SPP_N * PIX_PER_BLK * 2];

    const int tid  = threadIdx.x;
    const int pix  = tid & (PIX_PER_BLK - 1);  // pixel within segment
    const int b    = tid >> 6;                 // batch 0..3
    const int xseg = blockIdx.x;               // 0..3
    const int y    = blockIdx.y;               // 0..255
    const int x    = xseg * PIX_PER_BLK + pix;

    // ---- Stage jitter tile via async DMA: 16 slices x 64 px x 8 B = 8192 B ----
    {
        const uint64_t jbase = (uint64_t)(uintptr_t)jitter
                             + (uint64_t)((uint32_t)y * (SENS * 2u * 4u))      // y * 2048
                             + (uint64_t)((uint32_t)xseg * PIX_PER_BLK * 8u);  // x0 * 8
        const uint32_t lbase = (uint32_t)(uintptr_t)&ljit[0];
#pragma unroll
        for (int j = 0; j < 2; ++j) {
            uint32_t q   = ((uint32_t)tid + (uint32_t)j * 256u) * 16u; // byte 0..8176
            uint32_t s   = q >> 9;       // spp slice (512 B each)
            uint32_t off = q & 511u;     // byte within slice
            async_b128_to_lds_nt(lbase + q,
                                 jbase + (uint64_t)s * (SENS * SENS * 2u * 4u) + off);
        }
    }
    wait_asynccnt0();   // each wave drains its own ASYNCcnt...
    __syncthreads();    // ...then all waves rendezvous before reading LDS

    const float tt    = t[0];
    const float s_t   = gfx_tanh(tt);
    const float inv_s = __builtin_amdgcn_rcpf(s_t);
    const float stepx = 2.0f / (float)SENS;
    const float stepy = 2.0f / (float)SENS;
    const float posx0 = -1.0f + (float)x * stepx;
    const float posy0 = -1.0f + (float)y * stepy;

    const float* __restrict__ p0 = img + (size_t)b * NCHAN * IMGD * IMGD;
    const float* __restrict__ p1 = p0 + (size_t)IMGD * IMGD;
    const float* __restrict__ p2 = p1 + (size_t)IMGD * IMGD;

    float acc0 = 0.0f, acc1 = 0.0f, acc2 = 0.0f, area = 0.0f;

#pragma unroll 4
    for (int s = 0; s < SPP_N; ++s) {
        const float jx = ljit[s * (PIX_PER_BLK * 2) + pix * 2 + 0];
        const float jy = ljit[s * (PIX_PER_BLK * 2) + pix * 2 + 1];

        // warp + Jacobian (hardware tanh)
        const float px  = posx0 + jx * stepx;
        const float py  = posy0 + jy * stepy;
        const float thx = gfx_tanh(tt * px);
        const float thy = gfx_tanh(tt * py);
        const float wpx = thx * inv_s;
        const float wpy = thy * inv_s;
        const float ddx = tt * (1.0f - thx * thx) * inv_s;
        const float ddy = tt * (1.0f - thy * thy) * inv_s;
        const float det = ddx * ddy;
        area += det;

        // grid_sample (border clamp, bilinear)
        float gx = (wpx + 1.0f) * 512.0f - 0.5f;
        float gy = (wpy + 1.0f) * 512.0f - 0.5f;
        gx = fminf(fmaxf(gx, 0.0f), (float)(IMGD - 1));
        gy = fminf(fmaxf(gy, 0.0f), (float)(IMGD - 1));
        const float x0f = floorf(gx);
        const float y0f = floorf(gy);
        const float wx  = gx - x0f;
        const float wy  = gy - y0f;
        const int x0 = (int)x0f;
        const int y0 = (int)y0f;
        int x1 = x0 + 1; if (x1 > IMGD - 1) x1 = IMGD - 1;
        int y1 = y0 + 1; if (y1 > IMGD - 1) y1 = IMGD - 1;

        const int i00 = y0 * IMGD + x0;
        const int i01 = y0 * IMGD + x1;
        const int i10 = y1 * IMGD + x0;
        const int i11 = y1 * IMGD + x1;

        // fold det into the bilinear weights (shared across the 3 channels)
        const float w00 = (1.0f - wx) * (1.0f - wy) * det;
        const float w01 = wx * (1.0f - wy) * det;
        const float w10 = (1.0f - wx) * wy * det;
        const float w11 = wx * wy * det;

        acc0 += w00 * p0[i00] + w01 * p0[i01] + w10 * p0[i10] + w11 * p0[i11];
        acc1 += w00 * p1[i00] + w01 * p1[i01] + w10 * p1[i10] + w11 * p1[i11];
        acc2 += w00 * p2[i00] + w01 * p2[i01] + w10 * p2[i10] + w11 * p2[i11];
    }

    // out = Σ det·sample / Σ det   (the 1/16 means cancel)
    const float inv_area = __builtin_amdgcn_rcpf(area);
    const size_t plane = (size_t)SENS * SENS;
    const size_t o = (size_t)b * NCHAN * plane + (size_t)y * SENS + (size_t)x;
    // Outputs are write-once, never re-read: NT stores keep L2 for the image.
    __builtin_nontemporal_store(acc0 * inv_area, &out[o]);
    __builtin_nontemporal_store(acc1 * inv_area, &out[o + plane]);
    __builtin_nontemporal_store(acc2 * inv_area, &out[o + 2 * plane]);
}

extern "C" void kernel_launch(void* const* d_in, const int* in_sizes, int n_in,
                              void* d_out, int out_size, void* d_ws, size_t ws_size,
                              hipStream_t stream) {
    (void)in_sizes; (void)n_in; (void)d_ws; (void)ws_size; (void)out_size;
    const float* img    = (const float*)d_in[0];
    const float* t      = (const float*)d_in[1];
    const float* jitter = (const float*)d_in[2];
    float* out          = (float*)d_out;

    dim3 grid(SENS / PIX_PER_BLK, SENS);  // (4, 256) blocks
    dim3 block(256);                      // 8 wave32s; tid = (batch<<6) | pixel
    foveated_kernel<<<grid, block, 0, stream>>>(img, t, jitter, out);
}